// plga_layer_28217935135482
// MI455X (gfx1250) — compile-verified
//
#include <hip/hip_runtime.h>
#include <hip/hip_bf16.h>

// Problem sizes (fixed by the reference)
#define BB 2
#define HH 16
#define SS 2048
#define DD 64
#define BH (BB*HH)

typedef float v2f __attribute__((ext_vector_type(2)));
typedef float v8f __attribute__((ext_vector_type(8)));

// D = A(16x4 f32) x B(4x16 f32) + C(16x16 f32), wave32 WMMA
static __device__ __forceinline__ v8f wmma4(v2f a, v2f b, v8f c) {
    return __builtin_amdgcn_wmma_f32_16x16x4_f32(
        /*neg_a=*/false, a, /*neg_b=*/false, b,
        /*c_mod=*/(short)0, c, /*reuse_a=*/false, /*reuse_b=*/false);
}

// ---------------------------------------------------------------------------
// Kernel 1: per (b,h): WA = W@A + b ; AW = WA^2*sigmoid(WA)+eps ;
//           Ap = AW^pw ; avAp = a_vec@Ap + ba.   (tiny: 32 blocks)
// Writes avAp to d_out row-major and to workspace TRANSPOSED (for b64 B-loads).
// ---------------------------------------------------------------------------
__global__ void k1_prepare(const float* __restrict__ A, const float* __restrict__ W,
                           const float* __restrict__ bmat, const float* __restrict__ ba,
                           const float* __restrict__ pw, const float* __restrict__ avec,
                           float* __restrict__ outAW, float* __restrict__ outAvAp,
                           float* __restrict__ wsAvApT) {
    __shared__ float Ap[64 * 64];
    const int bh = blockIdx.x;          // b*H + h
    const int h  = bh & (HH - 1);
    const float* Ab = A + (size_t)bh * 4096;
    const float* Wh = W + (size_t)h * 4096;

    for (int idx = threadIdx.x; idx < 4096; idx += blockDim.x) {
        const int f = idx >> 6, e = idx & 63;
        float acc = bmat[h * 4096 + idx];
        #pragma unroll 8
        for (int d = 0; d < 64; ++d)
            acc = fmaf(Wh[f * 64 + d], Ab[d * 64 + e], acc);
        const float sig = 1.0f / (1.0f + __expf(-acc));
        const float aw  = acc * acc * sig + 1e-9f;      // strictly > 0
        outAW[(size_t)bh * 4096 + idx] = aw;
        Ap[idx] = powf(aw, pw[h * 4096 + idx]);
    }
    __syncthreads();
    for (int idx = threadIdx.x; idx < 4096; idx += blockDim.x) {
        const int f = idx >> 6, e = idx & 63;
        float acc = ba[h * 4096 + idx];
        #pragma unroll 8
        for (int d = 0; d < 64; ++d)
            acc = fmaf(avec[h * 4096 + f * 64 + d], Ap[d * 64 + e], acc);
        outAvAp[(size_t)bh * 4096 + idx] = acc;          // [f][e] row-major (output)
        wsAvApT[(size_t)bh * 4096 + e * 64 + f] = acc;   // [e][f] transposed (B operand)
    }
}

// ---------------------------------------------------------------------------
// Kernel 1b: transpose Hv -> HvT [bh][64][S] so the E@Hv B-operand becomes
// contiguous v2f (global_load_b64). LDS-tiled, conflict-free (stride 65).
// ---------------------------------------------------------------------------
__global__ void k1b_transpose_v(const float* __restrict__ Hv, float* __restrict__ HvT) {
    __shared__ float tile[64 * 65];
    const int blk  = blockIdx.x;
    const int tblk = blk & 31;          // S/64 tiles per bh
    const int bh   = blk >> 5;
    const int t0   = tblk * 64;
    const float* src = Hv + (size_t)bh * SS * 64;
    float* dst = HvT + (size_t)bh * 64 * SS;

    #pragma unroll
    for (int k = 0; k < 16; ++k) {
        const int lin = k * 256 + threadIdx.x;
        const int r = lin >> 6, c = lin & 63;            // r = t-local, c = d
        tile[r * 65 + c] = src[(size_t)(t0 + r) * 64 + c];
    }
    __syncthreads();
    #pragma unroll
    for (int k = 0; k < 16; ++k) {
        const int lin = k * 256 + threadIdx.x;
        const int d = lin >> 6, tl = lin & 63;           // write rows of HvT
        dst[(size_t)d * SS + t0 + tl] = tile[tl * 65 + d];
    }
}

// ---------------------------------------------------------------------------
// Kernel 2: Q = Hin @ avAp  per (b,h).  WMMA f32 16x16x4.
// 4 waves/block, each wave owns a 16-row strip, full 64-col output.
// B operand read from transposed avAp -> b64 loads.
// ---------------------------------------------------------------------------
__global__ void k2_query(const float* __restrict__ Hin,
                         const float* __restrict__ avApT,
                         float* __restrict__ Q) {
    const int blk  = blockIdx.x;
    const int sblk = blk & 31;          // S/64 = 32 row-blocks per bh
    const int bh   = blk >> 5;
    const int wave = threadIdx.x >> 5;
    const int lane = threadIdx.x & 31;
    const int m  = lane & 15;
    const int hi = lane >> 4;
    const int row0 = sblk * 64 + wave * 16;

    const float* Hb = Hin   + (size_t)bh * SS * 64;
    const float* PT = avApT + (size_t)bh * 4096;    // [e][k]

    v2f a[16];                           // A strip: Hin[row][k..k+1], ISA A layout
    #pragma unroll
    for (int i = 0; i < 16; ++i)
        a[i] = *(const v2f*)(Hb + (size_t)(row0 + m) * 64 + 4 * i + 2 * hi);

    #pragma unroll
    for (int ct = 0; ct < 4; ++ct) {
        v8f acc = {};
        #pragma unroll
        for (int i = 0; i < 16; ++i) {
            // B[k][n] = avAp[k][ct*16+n] = PT[(ct*16+n)*64 + k]
            v2f bv = *(const v2f*)(PT + (size_t)(ct * 16 + m) * 64 + 4 * i + 2 * hi);
            acc = wmma4(a[i], bv, acc);
        }
        #pragma unroll
        for (int r = 0; r < 8; ++r)
            Q[(size_t)bh * SS * 64 + (size_t)(row0 + r + 8 * hi) * 64 + ct * 16 + m] = acc[r];
    }
}

// ---------------------------------------------------------------------------
// Kernel 3 (pass A): Ep = Q @ Hk^T / 8 (full S x S, written out NT) + online
// masked softmax statistics (row max, 1/rowsum) into workspace.
// 8 waves/block, each wave = 16-row strip, loops over all 128 col tiles.
// ---------------------------------------------------------------------------
__global__ void k3_scores(const float* __restrict__ Q, const float* __restrict__ Hk,
                          float* __restrict__ Ep,
                          float* __restrict__ rowmax, float* __restrict__ rowrinv) {
    const int blk  = blockIdx.x;
    const int sblk = blk & 15;          // S/128 = 16 row-blocks per bh
    const int bh   = blk >> 4;
    const int wave = threadIdx.x >> 5;
    const int lane = threadIdx.x & 31;
    const int m  = lane & 15;
    const int hi = lane >> 4;
    const int row0 = sblk * 128 + wave * 16;

    const float* Qb = Q  + (size_t)bh * SS * 64;
    const float* Kb = Hk + (size_t)bh * SS * 64;
    float* Eb = Ep + (size_t)bh * SS * SS;

    v2f a[16];
    #pragma unroll
    for (int i = 0; i < 16; ++i)
        a[i] = *(const v2f*)(Qb + (size_t)(row0 + m) * 64 + 4 * i + 2 * hi);

    float mrun[8], lrun[8];
    #pragma unroll
    for (int r = 0; r < 8; ++r) { mrun[r] = -3.0e38f; lrun[r] = 0.0f; }

    for (int t0 = 0; t0 < SS; t0 += 16) {
        v8f c = {};
        #pragma unroll
        for (int i = 0; i < 16; ++i) {
            // B[k][n] = Hk[t0+n][k]  (K^T), lane n = m
            v2f bv = *(const v2f*)(Kb + (size_t)(t0 + m) * 64 + 4 * i + 2 * hi);
            c = wmma4(a[i], bv, c);
        }
        const int col = t0 + m;
        #pragma unroll
        for (int r = 0; r < 8; ++r) {
            const float x   = c[r] * 0.125f;          // 1/sqrt(64)
            const int   row = row0 + r + 8 * hi;
            // raw Ep (unmasked) is an output; streamed, re-read once by k4 -> NT
            __builtin_nontemporal_store(x, Eb + (size_t)row * SS + col);
            if (col <= row) {                         // causal-valid -> softmax stats
                const float mn = fmaxf(mrun[r], x);
                lrun[r] = lrun[r] * __expf(mrun[r] - mn) + __expf(x - mn);
                mrun[r] = mn;
            }
        }
    }
    // combine (max, sum) across the 16 lanes of each half-group
    #pragma unroll
    for (int r = 0; r < 8; ++r) {
        float mm = mrun[r], ll = lrun[r];
        #pragma unroll
        for (int off = 1; off < 16; off <<= 1) {
            const float m2 = __shfl_xor(mm, off, 32);
            const float l2 = __shfl_xor(ll, off, 32);
            const float mn = fmaxf(mm, m2);
            ll = ll * __expf(mm - mn) + l2 * __expf(m2 - mn);
            mm = mn;
        }
        if (m == 0) {
            const int row = row0 + r + 8 * hi;
            rowmax[(size_t)bh * SS + row] = mm;
            rowrinv[(size_t)bh * SS + row] = 1.0f / ll;
        }
    }
}

// ---------------------------------------------------------------------------
// Kernel 4 (pass B): E = exp(Ep - max)*rinv (0 above diagonal), write E (NT),
// fused Hout = E @ Hv via WMMA (E tiles land directly in A-operand layout).
// B operand from transposed Hv -> b64 loads.
// ---------------------------------------------------------------------------
__global__ void k4_softmax_out(const float* __restrict__ EpIn, const float* __restrict__ HvT,
                               const float* __restrict__ rowmax, const float* __restrict__ rowrinv,
                               float* __restrict__ Eout, float* __restrict__ Hout) {
    const int blk  = blockIdx.x;
    const int sblk = blk & 15;
    const int bh   = blk >> 4;
    const int wave = threadIdx.x >> 5;
    const int lane = threadIdx.x & 31;
    const int m  = lane & 15;
    const int hi = lane >> 4;
    const int row0 = sblk * 128 + wave * 16;
    const int arow = row0 + m;          // this lane's row for A-layout ops

    const float* Epb = EpIn + (size_t)bh * SS * SS;
    const float* VT  = HvT  + (size_t)bh * 64 * SS;   // [d][t]
    float* Eb = Eout + (size_t)bh * SS * SS;

    const float mrow = rowmax[(size_t)bh * SS + arow];
    const float rrow = rowrinv[(size_t)bh * SS + arow];

    v8f c[4] = {};
    const int ntile_active = (row0 >> 4) + 1;   // tiles touching the causal region

    for (int tt = 0; tt < SS / 16; ++tt) {
        const int t0 = tt * 16;
        if (tt < ntile_active) {
            v2f e[4];
            #pragma unroll
            for (int i = 0; i < 4; ++i) {
                const int col0 = t0 + 4 * i + 2 * hi;
                const v2f ep = __builtin_nontemporal_load(
                                   (const v2f*)(Epb + (size_t)arow * SS + col0));
                e[i].x = (col0     <= arow) ? __expf(ep.x - mrow) * rrow : 0.0f;
                e[i].y = (col0 + 1 <= arow) ? __expf(ep.y - mrow) * rrow : 0.0f;
                __builtin_nontemporal_store(e[i], (v2f*)(Eb + (size_t)arow * SS + col0));
            }
            #pragma unroll
            for (int ct = 0; ct < 4; ++ct) {
                #pragma unroll
                for (int k = 0; k < 4; ++k) {
                    const int trow = t0 + 4 * k + 2 * hi;   // B[k][n] = Hv[t0+k][ct*16+n]
                    v2f bv = *(const v2f*)(VT + (size_t)(ct * 16 + m) * SS + trow);
                    c[ct] = wmma4(e[k], bv, c[ct]);
                }
            }
        } else {
            v2f z; z.x = 0.0f; z.y = 0.0f;    // fully-masked tile: E = 0
            #pragma unroll
            for (int i = 0; i < 4; ++i)
                __builtin_nontemporal_store(
                    z, (v2f*)(Eb + (size_t)arow * SS + t0 + 4 * i + 2 * hi));
        }
    }
    #pragma unroll
    for (int ct = 0; ct < 4; ++ct)
        #pragma unroll
        for (int r = 0; r < 8; ++r)
            Hout[(size_t)bh * SS * 64 + (size_t)(row0 + r + 8 * hi) * 64 + ct * 16 + m] = c[ct][r];
}

// ---------------------------------------------------------------------------
// Output layout (floats), in reference return order:
//   Hout, E, AW, pw, a_vec, ba, avAp, Ep
// ---------------------------------------------------------------------------
#define N_HOUT ((size_t)BB * HH * SS * DD)          // 4,194,304
#define N_SS   ((size_t)BB * HH * SS * SS)          // 134,217,728
#define N_FD   ((size_t)BB * HH * 64 * 64)          // 131,072
#define N_HFD  ((size_t)HH * 64 * 64)               // 65,536

#define OFF_HOUT ((size_t)0)
#define OFF_E    (OFF_HOUT + N_HOUT)
#define OFF_AW   (OFF_E + N_SS)
#define OFF_PW   (OFF_AW + N_FD)
#define OFF_AVEC (OFF_PW + N_HFD)
#define OFF_BA   (OFF_AVEC + N_HFD)
#define OFF_AVAP (OFF_BA + N_HFD)
#define OFF_EP   (OFF_AVAP + N_FD)

extern "C" void kernel_launch(void* const* d_in, const int* in_sizes, int n_in,
                              void* d_out, int out_size, void* d_ws, size_t ws_size,
                              hipStream_t stream) {
    const float* Hin  = (const float*)d_in[0];
    const float* Hk   = (const float*)d_in[1];
    const float* Hv   = (const float*)d_in[2];
    const float* A    = (const float*)d_in[3];
    // d_in[4] = mask (causality is applied analytically)
    const float* W    = (const float*)d_in[5];
    const float* bmat = (const float*)d_in[6];
    const float* ba   = (const float*)d_in[7];
    const float* pw   = (const float*)d_in[8];
    const float* avec = (const float*)d_in[9];

    float* out = (float*)d_out;
    float* ws  = (float*)d_ws;
    float* wsAvApT = ws;                        // 131,072 floats
    float* wsQ     = wsAvApT + N_FD;            // 4,194,304 floats
    float* wsMax   = wsQ + N_HOUT;              // 65,536 floats
    float* wsRinv  = wsMax + (size_t)BB * HH * SS;
    float* wsHvT   = wsRinv + (size_t)BB * HH * SS;  // 4,194,304 floats

    // pass-through outputs
    hipMemcpyAsync(out + OFF_PW,   pw,   N_HFD * sizeof(float), hipMemcpyDeviceToDevice, stream);
    hipMemcpyAsync(out + OFF_AVEC, avec, N_HFD * sizeof(float), hipMemcpyDeviceToDevice, stream);
    hipMemcpyAsync(out + OFF_BA,   ba,   N_HFD * sizeof(float), hipMemcpyDeviceToDevice, stream);

    k1_prepare<<<BH, 256, 0, stream>>>(A, W, bmat, ba, pw, avec,
                                       out + OFF_AW, out + OFF_AVAP, wsAvApT);
    k1b_transpose_v<<<BH * (SS / 64), 256, 0, stream>>>(Hv, wsHvT);
    k2_query<<<BH * (SS / 64), 128, 0, stream>>>(Hin, wsAvApT, wsQ);
    k3_scores<<<BH * (SS / 128), 256, 0, stream>>>(wsQ, Hk, out + OFF_EP, wsMax, wsRinv);
    k4_softmax_out<<<BH * (SS / 128), 256, 0, stream>>>(out + OFF_EP, wsHvT, wsMax, wsRinv,
                                                        out + OFF_E, out + OFF_HOUT);
}